// MainAttention_40948218200277
// MI455X (gfx1250) — compile-verified
//
#include <hip/hip_runtime.h>
#include <hip/hip_bf16.h>

typedef __attribute__((ext_vector_type(2))) float f32x2;
typedef __attribute__((ext_vector_type(8))) float f32x8;

#define Bq 16
#define Nq 785
#define Cq 768
#define Hq 12
#define HDq 64
#define NPAD 800
#define SCALE 0.125f
#define LEFT 549
#define REG (Bq*Hq*NPAD*HDq)          // 9,830,400 floats per q/k/v matrix
#define QTILES 50                      // ceil(785/16)

// d_out layout (floats): out | idx | index | cls_attn
#define IDX_OFF   (Bq*Nq*Cq)                     // 9,646,080
#define INDEX_OFF (IDX_OFF + Bq*LEFT)            // 9,654,864
#define CLS_OFF   (INDEX_OFF + Bq*LEFT*Cq)       // 16,400,976

__device__ __forceinline__ f32x8 wmma4(f32x2 a, f32x2 b, f32x8 c) {
    // D = A(16x4) * B(4x16) + C(16x16), fp32
    return __builtin_amdgcn_wmma_f32_16x16x4_f32(false, a, false, b, (short)0, c, false, false);
}

// ---------------- zero pad rows 785..799 of q/k/v ----------------
__global__ __launch_bounds__(256) void zeropad_kernel(float* qkv) {
    int t = blockIdx.x * 256 + threadIdx.x;
    const int per = Bq*Hq*15*HDq;          // pad elems per matrix
    if (t >= 3*per) return;
    int which = t / per;
    int r = t % per;
    int d = r & 63;
    int row = (r >> 6) % 15;
    int bh = r / (15*64);
    qkv[which*REG + (bh*NPAD + Nq + row)*HDq + d] = 0.f;
}

// ---------------- generic WMMA GEMM: Y = A @ W^T (+bias) ----------------
// A: [M,768] row-major, W: [Nout,768] row-major. One wave -> 16x64 tile.
// mode 0: scatter into qkv [3,B,H,NPAD,64]; mode 1: plain [M,768] + bias.
__global__ __launch_bounds__(256) void gemm_kernel(const float* __restrict__ A,
                                                   const float* __restrict__ W,
                                                   const float* __restrict__ bias,
                                                   float* __restrict__ Yp,
                                                   float* __restrict__ QKV,
                                                   int Mtiles, int Ntiles64, int mode) {
    int wid = (blockIdx.x * blockDim.x + threadIdx.x) >> 5;
    if (wid >= Mtiles * Ntiles64) return;
    int nt64 = wid % Ntiles64;
    int mt   = wid / Ntiles64;
    int lane = threadIdx.x & 31;
    int hl = lane >> 4, l16 = lane & 15;

    const float* ab = A + (size_t)(mt*16 + l16)*768 + 2*hl;
    const float* wb0 = W + (size_t)(nt64*64 +  0 + l16)*768 + 2*hl;
    const float* wb1 = W + (size_t)(nt64*64 + 16 + l16)*768 + 2*hl;
    const float* wb2 = W + (size_t)(nt64*64 + 32 + l16)*768 + 2*hl;
    const float* wb3 = W + (size_t)(nt64*64 + 48 + l16)*768 + 2*hl;

    f32x8 acc[4];
    acc[0] = 0; acc[1] = 0; acc[2] = 0; acc[3] = 0;

    #pragma unroll 4
    for (int ks = 0; ks < 192; ++ks) {
        f32x2 a = *(const f32x2*)(ab + ks*4);
        f32x2 b0 = *(const f32x2*)(wb0 + ks*4);
        f32x2 b1 = *(const f32x2*)(wb1 + ks*4);
        f32x2 b2 = *(const f32x2*)(wb2 + ks*4);
        f32x2 b3 = *(const f32x2*)(wb3 + ks*4);
        acc[0] = wmma4(a, b0, acc[0]);
        acc[1] = wmma4(a, b1, acc[1]);
        acc[2] = wmma4(a, b2, acc[2]);
        acc[3] = wmma4(a, b3, acc[3]);
    }

    if (mode == 0) {
        #pragma unroll
        for (int c = 0; c < 4; ++c)
            #pragma unroll
            for (int w = 0; w < 8; ++w) {
                int mg = mt*16 + w + 8*hl;
                int ng = nt64*64 + c*16 + l16;
                int which = ng / 768;
                int cc = ng % 768;
                int hh = cc >> 6, d = cc & 63;
                int bb = mg / Nq, seq = mg % Nq;
                QKV[which*REG + ((bb*Hq + hh)*NPAD + seq)*HDq + d] = acc[c][w];
            }
    } else {
        #pragma unroll
        for (int c = 0; c < 4; ++c)
            #pragma unroll
            for (int w = 0; w < 8; ++w) {
                int mg = mt*16 + w + 8*hl;
                int ng = nt64*64 + c*16 + l16;
                Yp[(size_t)mg*768 + ng] = acc[c][w] + bias[ng];
            }
    }
}

// ---------------- flash attention: one wave per (b,h,16-query tile) ----------------
// Computes S^T = K*Q^T (so softmax stats are per-lane), then O^T = V^T * P^T.
__global__ __launch_bounds__(256) void attn_kernel(const float* __restrict__ Q,
                                                   const float* __restrict__ K,
                                                   const float* __restrict__ V,
                                                   float* __restrict__ AO) {
    int wid = (blockIdx.x * blockDim.x + threadIdx.x) >> 5;
    if (wid >= Bq*Hq*QTILES) return;
    int qt = wid % QTILES;
    int h  = (wid / QTILES) % Hq;
    int b  = wid / (QTILES*Hq);
    int lane = threadIdx.x & 31;
    int hl = lane >> 4, l16 = lane & 15;
    int bh = b*Hq + h;

    // Q^T B-fragments (pre-scaled): lane holds Q[q0+l16][k..k+1], k = 4*ks + 2*hl
    const float* qbase = Q + (size_t)(bh*NPAD + qt*16 + l16)*HDq + 2*hl;
    f32x2 qf[16];
    #pragma unroll
    for (int ks = 0; ks < 16; ++ks) {
        f32x2 t = *(const f32x2*)(qbase + ks*4);
        qf[ks] = t * SCALE;
    }

    float run_max = -INFINITY, run_sum = 0.f;
    f32x8 o[4];
    o[0] = 0; o[1] = 0; o[2] = 0; o[3] = 0;

    for (int jt = 0; jt < QTILES; ++jt) {
        int j0 = jt * 16;
        const float* kbase = K + (size_t)(bh*NPAD + j0 + l16)*HDq + 2*hl;
        f32x8 s = 0;
        #pragma unroll
        for (int ks = 0; ks < 16; ++ks) {
            f32x2 kf = *(const f32x2*)(kbase + ks*4);
            s = wmma4(kf, qf[ks], s);   // S^T tile: M=j, N=q
        }

        // mask invalid keys, per-lane tile max (all 8 values share one query q=l16)
        float p[8];
        float tm = -INFINITY;
        #pragma unroll
        for (int w = 0; w < 8; ++w) {
            int jl = w + 8*hl;
            float sv = (j0 + jl < Nq) ? s[w] : -INFINITY;
            p[w] = sv;
            tm = fmaxf(tm, sv);
        }
        tm = fmaxf(tm, __shfl_xor(tm, 16, 32));        // combine two j-halves
        float nm = fmaxf(run_max, tm);
        float corr = expf(run_max - nm);
        float ts = 0.f;
        #pragma unroll
        for (int w = 0; w < 8; ++w) { p[w] = expf(p[w] - nm); ts += p[w]; }
        ts += __shfl_xor(ts, 16, 32);
        run_sum = run_sum * corr + ts;
        run_max = nm;
        #pragma unroll
        for (int t = 0; t < 4; ++t) o[t] = o[t] * corr;  // per-lane q matches O^T columns

        // O^T += V^T(16x4 tiles) * P^T(4x16); P^T B-frags via xor-16 shuffles
        const float* vrow = V + (size_t)(bh*NPAD + j0)*HDq;
        #pragma unroll
        for (int kk = 0; kk < 4; ++kk) {
            f32x2 pb;
            if (kk < 2) {
                float shx = __shfl_xor(p[kk*4+2], 16, 32);
                float shy = __shfl_xor(p[kk*4+3], 16, 32);
                pb.x = hl ? shx : p[kk*4];
                pb.y = hl ? shy : p[kk*4+1];
            } else {
                int b8 = kk*4 - 8;
                float shx = __shfl_xor(p[b8],   16, 32);
                float shy = __shfl_xor(p[b8+1], 16, 32);
                pb.x = hl ? p[b8+2] : shx;
                pb.y = hl ? p[b8+3] : shy;
            }
            int tok = kk*4 + 2*hl;
            #pragma unroll
            for (int t = 0; t < 4; ++t) {
                const float* va = vrow + tok*HDq + t*16 + l16;
                f32x2 av;
                av.x = va[0];
                av.y = va[HDq];
                o[t] = wmma4(av, pb, o[t]);
            }
        }
    }

    float inv = 1.0f / run_sum;
    int qrow = qt*16 + l16;
    if (qrow < Nq) {
        float* dst = AO + (size_t)(b*Nq + qrow)*Cq + h*HDq;
        #pragma unroll
        for (int t = 0; t < 4; ++t)
            #pragma unroll
            for (int w = 0; w < 8; ++w)
                dst[t*16 + w + 8*hl] = o[t][w] * inv;
    }
}

// ---------------- CLS-attention row (mean over heads) ----------------
__global__ __launch_bounds__(256) void cls_kernel(const float* __restrict__ Q,
                                                  const float* __restrict__ K,
                                                  float* __restrict__ cls_out) {
    int b = blockIdx.x, tid = threadIdx.x;
    __shared__ float sc[Nq];
    __shared__ float accv[Nq-1];
    __shared__ float red[256];
    for (int j = tid; j < Nq-1; j += 256) accv[j] = 0.f;
    __syncthreads();
    for (int h = 0; h < Hq; ++h) {
        const float* q0 = Q + (size_t)((b*Hq + h)*NPAD)*HDq;
        const float* kb = K + (size_t)((b*Hq + h)*NPAD)*HDq;
        float lmax = -INFINITY;
        for (int j = tid; j < Nq; j += 256) {
            const float* kr = kb + j*HDq;
            float dot = 0.f;
            #pragma unroll 8
            for (int d = 0; d < HDq; ++d) dot += q0[d]*kr[d];
            dot *= SCALE;
            sc[j] = dot;
            lmax = fmaxf(lmax, dot);
        }
        red[tid] = lmax; __syncthreads();
        for (int off = 128; off > 0; off >>= 1) {
            if (tid < off) red[tid] = fmaxf(red[tid], red[tid+off]);
            __syncthreads();
        }
        float smax = red[0]; __syncthreads();
        float lsum = 0.f;
        for (int j = tid; j < Nq; j += 256) {
            float e = expf(sc[j] - smax);
            sc[j] = e;
            lsum += e;
        }
        red[tid] = lsum; __syncthreads();
        for (int off = 128; off > 0; off >>= 1) {
            if (tid < off) red[tid] += red[tid+off];
            __syncthreads();
        }
        float invs = 1.0f / (red[0] * (float)Hq);
        __syncthreads();
        for (int j = tid; j < Nq-1; j += 256) accv[j] += sc[j+1] * invs;
        __syncthreads();
    }
    for (int j = tid; j < Nq-1; j += 256) cls_out[b*(Nq-1) + j] = accv[j];
}

// ---------------- deterministic top-k via rank (matches stable top_k) ----------------
__global__ __launch_bounds__(256) void topk_kernel(const float* __restrict__ cls,
                                                   float* __restrict__ idx_out) {
    int b = blockIdx.x, tid = threadIdx.x;
    __shared__ float vals[Nq-1];
    for (int j = tid; j < Nq-1; j += 256) vals[j] = cls[b*(Nq-1) + j];
    __syncthreads();
    for (int j = tid; j < Nq-1; j += 256) {
        float val = vals[j];
        int rank = 0;
        for (int i = 0; i < Nq-1; ++i) {
            float vi = vals[i];
            rank += (vi > val) || (vi == val && i < j);
        }
        if (rank < LEFT) idx_out[b*LEFT + rank] = (float)j;
    }
}

__global__ __launch_bounds__(256) void index_kernel(const float* __restrict__ idx,
                                                    float* __restrict__ index_out) {
    int t = blockIdx.x * 256 + threadIdx.x;
    if (t >= Bq*LEFT*Cq) return;
    int r = (t / Cq) % LEFT;
    int b = t / (Cq*LEFT);
    index_out[t] = idx[b*LEFT + r];
}

extern "C" void kernel_launch(void* const* d_in, const int* in_sizes, int n_in,
                              void* d_out, int out_size, void* d_ws, size_t ws_size,
                              hipStream_t stream) {
    const float* x     = (const float*)d_in[0];   // [B,N,C]
    const float* Wqkv  = (const float*)d_in[1];   // [3C,C]
    const float* Wproj = (const float*)d_in[2];   // [C,C]
    const float* bproj = (const float*)d_in[3];   // [C]
    float* out = (float*)d_out;
    float* ws  = (float*)d_ws;

    float* Q  = ws;                 // [B,H,NPAD,64]
    float* K  = ws + (size_t)REG;
    float* V  = ws + (size_t)2*REG;
    float* AO = ws + (size_t)3*REG; // [B,N,C] attention output pre-projection

    // 1) zero qkv pad rows
    {
        int n = 3 * Bq*Hq*15*HDq;
        zeropad_kernel<<<(n + 255)/256, 256, 0, stream>>>(ws);
    }
    // 2) qkv = x @ Wqkv^T scattered into q/k/v
    {
        int mt = (Bq*Nq)/16, nt = (3*Cq)/64;            // 785 x 36
        int waves = mt*nt;
        gemm_kernel<<<(waves*32 + 255)/256, 256, 0, stream>>>(x, Wqkv, nullptr,
                                                              nullptr, ws, mt, nt, 0);
    }
    // 3) attention -> AO [B,N,C]
    {
        int waves = Bq*Hq*QTILES;                       // 9600
        attn_kernel<<<(waves*32 + 255)/256, 256, 0, stream>>>(Q, K, V, AO);
    }
    // 4) out = AO @ Wproj^T + bproj
    {
        int mt = (Bq*Nq)/16, nt = Cq/64;                // 785 x 12
        int waves = mt*nt;
        gemm_kernel<<<(waves*32 + 255)/256, 256, 0, stream>>>(AO, Wproj, bproj,
                                                              out, nullptr, mt, nt, 1);
    }
    // 5) cls_attn
    cls_kernel<<<Bq, 256, 0, stream>>>(Q, K, out + CLS_OFF);
    // 6) top-k indices
    topk_kernel<<<Bq, 256, 0, stream>>>(out + CLS_OFF, out + IDX_OFF);
    // 7) broadcast index [B,LEFT,C]
    {
        int n = Bq*LEFT*Cq;
        index_kernel<<<(n + 255)/256, 256, 0, stream>>>(out + IDX_OFF, out + INDEX_OFF);
    }
}